// Multihead_LSTM_25683904430310
// MI455X (gfx1250) — compile-verified
//
#include <hip/hip_runtime.h>
#include <cstdint>

// Problem constants (from reference): x:[64,2048,256] f32, per-channel 2-layer LSTM
// (hidden=1) + residual + LayerNorm over D. No GEMM -> no WMMA; CDNA5 paths used:
// async global->LDS streaming (ASYNCcnt), hardware v_tanh_f32, wave32 shuffles.
#define BB   64
#define TT   2048
#define DD   256
#define CH   16               // timesteps per chunk
#define NCH  (TT / CH)        // 128 chunks
#define LN_EPS 1e-5f

// ---- fast tanh: gfx1250 hardware v_tanh_f32 on every path ----
// Builtin preferred (compiler handles TRANS hazards); otherwise inline asm with
// an explicit TRANS32 delay per ISA 7.4 ("1 independent op after trans before
// output use") since the compiler can't see inside the asm blob.
__device__ __forceinline__ float ftanh(float x) {
#if __has_builtin(__builtin_amdgcn_tanhf)
  return __builtin_amdgcn_tanhf(x);
#elif __has_builtin(__builtin_amdgcn_tanh_f32)
  return __builtin_amdgcn_tanh_f32(x);
#else
  float r;
  asm("v_tanh_f32 %0, %1\n\t"
      "s_delay_alu instid0(TRANS32_DEP_1)"
      : "=v"(r) : "v"(x));
  return r;
#endif
}
__device__ __forceinline__ float fsig(float x) {
  // sigmoid(x) = 0.5*tanh(0.5x) + 0.5  (single trans op on the critical path)
  return fmaf(0.5f, ftanh(0.5f * x), 0.5f);
}

// ---- async copy of one 16KB x-chunk (4096 floats) into LDS ----
// Each of 256 threads issues 4x global_load_async_to_lds_b128 (16 floats).
// GVS mode: 64-bit SGPR base + 32-bit VGPR byte offset; LDS dest addr in VGPR.
__device__ __forceinline__ void async_chunk_load(const float* gbase,
                                                 uint32_t lds_base, int tid) {
  uint64_t sbase = (uint64_t)(uintptr_t)gbase;
#pragma unroll
  for (int k = 0; k < 4; ++k) {
    uint32_t voff  = (uint32_t)((tid * 4 + k * 1024) * 4);  // bytes
    uint32_t ldsa  = lds_base + voff;                       // same linear layout
    asm volatile("global_load_async_to_lds_b128 %0, %1, %2"
                 :: "v"(ldsa), "v"(voff), "s"(sbase)
                 : "memory");
  }
}

__global__ __launch_bounds__(DD) void mh_lstm_ln_kernel(
    const float* __restrict__ x,
    const float* __restrict__ w_ih0, const float* __restrict__ w_hh0,
    const float* __restrict__ b_ih0, const float* __restrict__ b_hh0,
    const float* __restrict__ w_ih1, const float* __restrict__ w_hh1,
    const float* __restrict__ b_ih1, const float* __restrict__ b_hh1,
    const float* __restrict__ gamma, const float* __restrict__ beta,
    float* __restrict__ out)
{
  __shared__ __align__(16) float xbuf[2][CH * DD];   // 2 x 16KB (double buffer)
  __shared__ __align__(16) float rbuf[CH * DD];      // 16KB residual chunk

  const int tid  = threadIdx.x;          // channel d for LSTM phase
  const int b    = blockIdx.x;           // batch
  const int lane = tid & 31;
  const int wv   = tid >> 5;             // wave id 0..7

  // Per-channel scalar weights in registers. Layout [D,4], gate order i,f,g,o.
  float wi0[4], wh0[4], bs0[4], wi1[4], wh1[4], bs1[4];
#pragma unroll
  for (int g = 0; g < 4; ++g) {
    wi0[g] = w_ih0[tid * 4 + g];
    wh0[g] = w_hh0[tid * 4 + g];
    bs0[g] = b_ih0[tid * 4 + g] + b_hh0[tid * 4 + g];
    wi1[g] = w_ih1[tid * 4 + g];
    wh1[g] = w_hh1[tid * 4 + g];
    bs1[g] = b_ih1[tid * 4 + g] + b_hh1[tid * 4 + g];
  }
  // gamma/beta for the channels this lane normalizes (lane + 32k)
  float lgam[8], lbet[8];
#pragma unroll
  for (int k = 0; k < 8; ++k) {
    lgam[k] = gamma[lane + 32 * k];
    lbet[k] = beta[lane + 32 * k];
  }

  float h0 = 0.f, c0 = 0.f, h1 = 0.f, c1 = 0.f;

  const float* xrow = x   + (size_t)b * TT * DD;
  float*       orow = out + (size_t)b * TT * DD;

  // Prefetch chunk 0 into buffer 0.
  async_chunk_load(xrow, (uint32_t)(uintptr_t)&xbuf[0][0], tid);

  int cur = 0;
  for (int c = 0; c < NCH; ++c) {
    // Kick off next chunk's async fetch into the other buffer, then wait for
    // the current chunk (async loads complete in order; each wave waits for
    // its own ops, cross-wave visibility via the barrier below).
    if (c + 1 < NCH) {
      async_chunk_load(xrow + (size_t)(c + 1) * CH * DD,
                       (uint32_t)(uintptr_t)&xbuf[cur ^ 1][0], tid);
      asm volatile("s_wait_asynccnt 0x4" ::: "memory");
    } else {
      asm volatile("s_wait_asynccnt 0x0" ::: "memory");
    }
    __syncthreads();

    // ---- LSTM phase: 16 sequential steps, register-resident recurrence ----
#pragma unroll 4
    for (int tl = 0; tl < CH; ++tl) {
      float xv = xbuf[cur][tl * DD + tid];
      // layer 0
      float zi = fmaf(h0, wh0[0], fmaf(xv, wi0[0], bs0[0]));
      float zf = fmaf(h0, wh0[1], fmaf(xv, wi0[1], bs0[1]));
      float zg = fmaf(h0, wh0[2], fmaf(xv, wi0[2], bs0[2]));
      float zo = fmaf(h0, wh0[3], fmaf(xv, wi0[3], bs0[3]));
      float gi = fsig(zi), gf = fsig(zf), gc = ftanh(zg), go = fsig(zo);
      c0 = fmaf(gf, c0, gi * gc);
      h0 = go * ftanh(c0);
      // layer 1 (input = h0)
      zi = fmaf(h1, wh1[0], fmaf(h0, wi1[0], bs1[0]));
      zf = fmaf(h1, wh1[1], fmaf(h0, wi1[1], bs1[1]));
      zg = fmaf(h1, wh1[2], fmaf(h0, wi1[2], bs1[2]));
      zo = fmaf(h1, wh1[3], fmaf(h0, wi1[3], bs1[3]));
      gi = fsig(zi); gf = fsig(zf); gc = ftanh(zg); go = fsig(zo);
      c1 = fmaf(gf, c1, gi * gc);
      h1 = go * ftanh(c1);
      rbuf[tl * DD + tid] = xv + h1;          // residual, conflict-free stride-1
    }
    __syncthreads();

    // ---- LayerNorm phase: each wave normalizes 2 full rows (D=256) ----
#pragma unroll
    for (int rr = 0; rr < 2; ++rr) {
      const int row = wv * 2 + rr;
      float v[8];
      float s = 0.f, q = 0.f;
#pragma unroll
      for (int k = 0; k < 8; ++k) {
        v[k] = rbuf[row * DD + lane + 32 * k];
        s += v[k];
        q  = fmaf(v[k], v[k], q);
      }
#pragma unroll
      for (int m = 16; m >= 1; m >>= 1) {     // wave32 butterfly reduction
        s += __shfl_xor(s, m, 32);
        q += __shfl_xor(q, m, 32);
      }
      const float mean = s * (1.0f / DD);
      const float var  = fmaf(-mean, mean, q * (1.0f / DD));
      const float rstd = rsqrtf(var + LN_EPS);
      float* op = orow + ((size_t)c * CH + row) * DD;
#pragma unroll
      for (int k = 0; k < 8; ++k) {
        op[lane + 32 * k] = fmaf((v[k] - mean) * rstd, lgam[k], lbet[k]);
      }
    }
    __syncthreads();   // protect rbuf / xbuf[cur^1] before next chunk
    cur ^= 1;
  }
}

extern "C" void kernel_launch(void* const* d_in, const int* in_sizes, int n_in,
                              void* d_out, int out_size, void* d_ws, size_t ws_size,
                              hipStream_t stream) {
  (void)in_sizes; (void)n_in; (void)out_size; (void)d_ws; (void)ws_size;
  const float* x     = (const float*)d_in[0];
  const float* w_ih0 = (const float*)d_in[1];
  const float* w_hh0 = (const float*)d_in[2];
  const float* b_ih0 = (const float*)d_in[3];
  const float* b_hh0 = (const float*)d_in[4];
  const float* w_ih1 = (const float*)d_in[5];
  const float* w_hh1 = (const float*)d_in[6];
  const float* b_ih1 = (const float*)d_in[7];
  const float* b_hh1 = (const float*)d_in[8];
  const float* gamma = (const float*)d_in[9];
  const float* beta  = (const float*)d_in[10];
  float* out = (float*)d_out;

  mh_lstm_ln_kernel<<<dim3(BB), dim3(DD), 0, stream>>>(
      x, w_ih0, w_hh0, b_ih0, b_hh0, w_ih1, w_hh1, b_ih1, b_hh1,
      gamma, beta, out);
}